// LearnedPooler_14224931684836
// MI455X (gfx1250) — compile-verified
//
#include <hip/hip_runtime.h>
#include <math.h>

// LearnedPooler for MI455X (gfx1250, wave32).
// h: (B=16, L=4096, W=1024) fp32; mask: (B,L) bool (1 byte each, True = masked);
// queries: (Q=4, W) fp32; out: (B, Q*W) fp32.
//
// Roofline: 268 MB of h, ~1.1 GFLOP -> pure HBM-bandwidth bound (~2 FLOP/B).
// Fused online-softmax so h is streamed from HBM exactly once (~11.5 us floor);
// the second read of each 1 MB chunk in Phase C hits L2 (192 MB).
// Scores use V_WMMA_F32_16X16X4_F32 (fp32-exact tensor path).

typedef __attribute__((ext_vector_type(2))) float v2f;
typedef __attribute__((ext_vector_type(4))) float v4f;
typedef __attribute__((ext_vector_type(8))) float v8f;

#define B_       16
#define L_       4096
#define W_       1024
#define Q_       4
#define CHUNKS   16
#define LC       (L_ / CHUNKS)   // 256 rows per chunk
#define NTHREADS 256
#define NWAVES   8
#define SCALE    0.03125f        // 1024^-0.5
#define NEG_BIG  -1e30f

__global__ __launch_bounds__(NTHREADS)
void pool_partial(const float* __restrict__ h,
                  const unsigned char* __restrict__ mask,
                  const float* __restrict__ queries,
                  float* __restrict__ ws_hdr,   // [B*CHUNKS][8]: m[0..3], s[0..3]
                  float* __restrict__ ws_acc)   // [B*CHUNKS][Q][W]
{
    __shared__ float qpad[5 * W_];      // 20 KB: rows 0..3 = queries, row 4 = zeros (pad)
    __shared__ float sc[LC * Q_];       // 4 KB: scores, then softmax probabilities
    __shared__ float red[NTHREADS];     // reduction scratch
    __shared__ float mq[Q_];

    const int tid = threadIdx.x;
    const int wg  = blockIdx.x;               // b*CHUNKS + c
    const int b   = wg / CHUNKS;
    const int c   = wg % CHUNKS;
    const int rowbase = c * LC;
    const float* hb = h + (size_t)b * L_ * W_;
    const unsigned char* mb = mask + (size_t)b * L_;

    // Stage queries into LDS, zero-padded to 5 "rows" so all 16 B-matrix
    // columns have a valid LDS source (WMMA needs EXEC all-ones; no predication).
    for (int i = tid; i < Q_ * W_; i += NTHREADS) qpad[i] = queries[i];
    for (int i = tid; i < W_;      i += NTHREADS) qpad[4 * W_ + i] = 0.0f;
    __syncthreads();

    // ---------------- Phase A: scores via V_WMMA_F32_16X16X4_F32 -------------
    // D[m][n] = sum_k A[m][k]*B[k][n], A[m][k] = h[row m][w=k], B[k][n] = q[n][w=k].
    // f32 A 16x4 layout: lane i -> row M = i&15, K-pair offset = (i>>4)*2.
    // f32 B 4x16 layout mirrors it: lane i -> col N = i&15, same K-pair offset.
    const int lane = tid & 31;
    const int wave = tid >> 5;
    const int l15  = lane & 15;
    const int koff = (lane >> 4) * 2;
    const int qsel = (l15 < Q_) ? l15 : 4;            // pad row -> zeros
    const float* qrow = qpad + qsel * W_ + koff;

    for (int g = wave; g < LC / 16; g += NWAVES) {    // 2 row-groups per wave
        const float* arow = hb + (size_t)(rowbase + g * 16 + l15) * W_ + koff;
        v8f acc = {};
#pragma unroll 4
        for (int k = 0; k < W_; k += 4) {
            v2f a, bb;
            a.x  = arow[k];  a.y  = arow[k + 1];
            bb.x = qrow[k];  bb.y = qrow[k + 1];
            acc = __builtin_amdgcn_wmma_f32_16x16x4_f32(
                    false, a, false, bb, (short)0, acc, false, false);
        }
        // D layout: lanes 0-15 hold M=r (VGPR r), lanes 16-31 hold M=8+r; N = lane&15.
        if (l15 < Q_) {
            const int mbase = g * 16 + ((lane >> 4) << 3);
#pragma unroll
            for (int r = 0; r < 8; ++r) {
                const int l = mbase + r;
                float v = acc[r] * SCALE;
                if (mb[rowbase + l]) v = NEG_BIG;     // masked -> large negative
                sc[l * Q_ + l15] = v;
            }
        }
    }
    __syncthreads();

    // ---------------- Phase B: chunk-local softmax (max, exp, sum) ----------
    {
        const int q = tid & 3;
        const int rb = tid >> 2;                      // 0..63
        float pm = NEG_BIG;
#pragma unroll
        for (int j = 0; j < LC / 64; ++j)
            pm = fmaxf(pm, sc[(rb + j * 64) * Q_ + q]);
        red[tid] = pm;
    }
    __syncthreads();
    if (tid < Q_) {
        float pm = NEG_BIG;
        for (int i = 0; i < 64; ++i) pm = fmaxf(pm, red[i * 4 + tid]);
        mq[tid] = pm;
    }
    __syncthreads();
    {
        const int q = tid & 3;
        const int rb = tid >> 2;
        const float m = mq[q];
        float ps = 0.f;
#pragma unroll
        for (int j = 0; j < LC / 64; ++j) {
            const int l = rb + j * 64;
            float p = __expf(sc[l * Q_ + q] - m);
            sc[l * Q_ + q] = p;
            ps += p;
        }
        red[tid] = ps;
    }
    __syncthreads();
    if (tid < Q_) {
        float s = 0.f;
        for (int i = 0; i < 64; ++i) s += red[i * 4 + tid];
        ws_hdr[wg * 8 + tid]     = mq[tid];
        ws_hdr[wg * 8 + 4 + tid] = s;
    }
    __syncthreads();

    // ---------------- Phase C: acc[q][w] = sum_l p[l][q] * h[l][w] ----------
    // Thread t owns w = 4t..4t+3 -> each wave issues one fully-coalesced
    // 512B global_load_b128 per row; rows come from L2 (just streamed).
    {
        const int w0 = tid * 4;
        v4f a0 = {}, a1 = {}, a2 = {}, a3 = {};
        const float* hrow = hb + (size_t)rowbase * W_ + w0;
        for (int l = 0; l < LC; ++l) {
            const v4f hv = *(const v4f*)(hrow + (size_t)l * W_);
            const v4f pv = *(const v4f*)(&sc[l * Q_]);   // LDS broadcast
            a0 += pv.x * hv;
            a1 += pv.y * hv;
            a2 += pv.z * hv;
            a3 += pv.w * hv;
        }
        float* ao = ws_acc + (size_t)wg * (Q_ * W_);
        *(v4f*)(ao + 0 * W_ + w0) = a0;
        *(v4f*)(ao + 1 * W_ + w0) = a1;
        *(v4f*)(ao + 2 * W_ + w0) = a2;
        *(v4f*)(ao + 3 * W_ + w0) = a3;
    }
}

__global__ __launch_bounds__(NTHREADS)
void pool_combine(const float* __restrict__ ws_hdr,
                  const float* __restrict__ ws_acc,
                  float* __restrict__ out)
{
    const int b   = blockIdx.x;
    const int tid = threadIdx.x;

    // Global log-sum-exp merge of the 16 chunk partials (computed redundantly
    // per thread; 64 tiny reads, avoids a barrier).
    float M[Q_], Dn[Q_];
#pragma unroll
    for (int q = 0; q < Q_; ++q) {
        float m = NEG_BIG;
        for (int c = 0; c < CHUNKS; ++c)
            m = fmaxf(m, ws_hdr[(b * CHUNKS + c) * 8 + q]);
        float d = 0.f;
        for (int c = 0; c < CHUNKS; ++c) {
            const float* hd = ws_hdr + (b * CHUNKS + c) * 8;
            d += __expf(hd[q] - m) * hd[4 + q];
        }
        M[q] = m;
        Dn[q] = d;
    }

    const int w0 = tid * 4;
#pragma unroll
    for (int q = 0; q < Q_; ++q) {
        v4f o = {};
        for (int c = 0; c < CHUNKS; ++c) {
            const int wg = b * CHUNKS + c;
            const float wt = __expf(ws_hdr[wg * 8 + q] - M[q]);
            const v4f av = *(const v4f*)(ws_acc + (size_t)wg * (Q_ * W_) + q * W_ + w0);
            o += wt * av;
        }
        const float inv = 1.0f / Dn[q];
        *(v4f*)(out + (size_t)b * (Q_ * W_) + q * W_ + w0) = o * inv;
    }
}

extern "C" void kernel_launch(void* const* d_in, const int* in_sizes, int n_in,
                              void* d_out, int out_size, void* d_ws, size_t ws_size,
                              hipStream_t stream) {
    const float*         h       = (const float*)d_in[0];
    const unsigned char* mask    = (const unsigned char*)d_in[1];  // jnp.bool_ = 1 byte
    const float*         queries = (const float*)d_in[2];
    float*               out     = (float*)d_out;

    // Workspace layout: headers (B*CHUNKS*8 floats = 8 KB) then partial
    // accumulators (B*CHUNKS*Q*W floats = 4 MB). Total ~4.2 MB.
    float* hdr = (float*)d_ws;
    float* acc = hdr + (size_t)B_ * CHUNKS * 8;

    pool_partial<<<B_ * CHUNKS, NTHREADS, 0, stream>>>(h, mask, queries, hdr, acc);
    pool_combine<<<B_,          NTHREADS, 0, stream>>>(hdr, acc, out);
}